// controller_35261681500897
// MI455X (gfx1250) — compile-verified
//
#include <hip/hip_runtime.h>
#include <math.h>

// Sequential 25-cell LSTM chain, hidden=3, input=1.
// Latency-bound: single wave32, one V_WMMA_F32_16X16X4_F32 per cell doing
//   gates(12) = W_hh(12x3 -> A 16x4) @ h(3 -> B col 0) + bias (folded into C).
// Gate broadcast via v_readlane (no LDS pipe), activations via native
// v_tanh_f32 when available (branch-free either way).

typedef __attribute__((ext_vector_type(2))) float v2f;
typedef __attribute__((ext_vector_type(8))) float v8f;

#define N_CELLS 25
#define HID     3
#define G4      12   // 4*HID

// ---- branch-free transcendentals on the serial critical path ---------------

__device__ __forceinline__ float rcp_fast(float x) {
    return __builtin_amdgcn_rcpf(x);           // v_rcp_f32
}

__device__ __forceinline__ float tanh_fast(float x) {
#if __has_builtin(__builtin_amdgcn_tanhf)
    return __builtin_amdgcn_tanhf(x);          // v_tanh_f32 (gfx1250)
#else
    // branch-free: tanh(x) = sign(x) * (1 - t)/(1 + t), t = exp(-2|x|)
    float t = __expf(-2.0f * fabsf(x));
    float r = (1.0f - t) * rcp_fast(1.0f + t);
    return copysignf(r, x);
#endif
}

__device__ __forceinline__ float sigmoid_fast(float x) {
#if __has_builtin(__builtin_amdgcn_tanhf)
    // sigma(x) = 0.5*tanh(x/2) + 0.5  -> single v_tanh_f32
    return fmaf(0.5f, __builtin_amdgcn_tanhf(0.5f * x), 0.5f);
#else
    return rcp_fast(1.0f + __expf(-x));
#endif
}

__device__ __forceinline__ float bcast(float v, int srcLane) {
    return __int_as_float(__builtin_amdgcn_readlane(__float_as_int(v), srcLane));
}

// ----------------------------------------------------------------------------

__global__ __launch_bounds__(32)
void lstm_chain_wmma_kernel(const float* __restrict__ state,   // (25,)
                            const float* __restrict__ W_ih,    // (25,12,1)
                            const float* __restrict__ W_hh,    // (25,12,3)
                            const float* __restrict__ b_ih,    // (25,12)
                            const float* __restrict__ b_hh,    // (25,12)
                            float* __restrict__ out)           // (25,3)
{
    const int lane  = threadIdx.x;      // 0..31 (wave32)
    const int m     = lane & 15;        // A-matrix row covered by this lane
    const int khalf = lane >> 4;        // 0 -> K={0,1}, 1 -> K={2,3}
    const int k0 = 2 * khalf;
    const int k1 = k0 + 1;

    float h0 = 0.f, h1 = 0.f, h2 = 0.f;   // hidden state (wave-uniform)
    float c0 = 0.f, c1 = 0.f, c2 = 0.f;   // cell state   (wave-uniform)

    for (int cell = 0; cell < N_CELLS; ++cell) {
        // Prefetch next cell's weights into cache while this cell computes.
        if (cell + 1 < N_CELLS) {
            __builtin_prefetch(W_hh + (cell + 1) * (G4 * HID), 0, 3);
            __builtin_prefetch(b_ih + (cell + 1) * G4, 0, 3);
        }

        const float x = state[cell];
        const float* Whh = W_hh + cell * (G4 * HID);   // 12x3 row-major

        // ---- A matrix (16x4 f32): lane m+16*khalf holds K=k0,k1; zero-pad.
        v2f a;
        a.x = (m < G4 && k0 < HID) ? Whh[m * HID + k0] : 0.f;
        a.y = (m < G4 && k1 < HID) ? Whh[m * HID + k1] : 0.f;

        // ---- B matrix (4x16 f32): only column 0 carries h, rest zero.
        //      Row K -> (b.x lanes0-15)=K0, (b.x lanes16-31)=K1,
        //               (b.y lanes0-15)=K2, (b.y lanes16-31)=K3.
        v2f b;
        b.x = (lane == 0) ? h0 : ((lane == 16) ? h1 : 0.f);
        b.y = (lane == 0) ? h2 : 0.f;

        // ---- C matrix: fused bias at D's (m,0) slots.
        //      D[m,0]: m=0..7 -> lane 0, vgpr m ; m=8..15 -> lane 16, vgpr m-8.
        v8f cacc = {};
        #pragma unroll
        for (int v = 0; v < 8; ++v) {
            int mm = (lane == 0) ? v : ((lane == 16) ? (8 + v) : 16);
            float bias = 0.f;
            if (mm < G4) {
                bias = W_ih[cell * G4 + mm] * x
                     + b_ih[cell * G4 + mm]
                     + b_hh[cell * G4 + mm];
            }
            cacc[v] = bias;
        }

        // ---- One matrix op per cell: gates = A*B + C  (exact fp32).
        v8f d = __builtin_amdgcn_wmma_f32_16x16x4_f32(
            /*neg_a=*/false, a, /*neg_b=*/false, b,
            /*c_mod=*/(short)0, cacc, /*reuse_a=*/false, /*reuse_b=*/false);

        // ---- Broadcast the 12 gate pre-activations via v_readlane (uniform,
        //      no LDS pipe, no DScnt waits).
        float g[G4];
        #pragma unroll
        for (int v = 0; v < 8; ++v) g[v]     = bcast(d[v], 0);
        #pragma unroll
        for (int v = 0; v < 4; ++v) g[8 + v] = bcast(d[v], 16);

        // Torch gate order [i, f, g, o], each of width HID=3.
        float ig0 = sigmoid_fast(g[0]),  ig1 = sigmoid_fast(g[1]),  ig2 = sigmoid_fast(g[2]);
        float fg0 = sigmoid_fast(g[3]),  fg1 = sigmoid_fast(g[4]),  fg2 = sigmoid_fast(g[5]);
        float gg0 = tanh_fast(g[6]),     gg1 = tanh_fast(g[7]),     gg2 = tanh_fast(g[8]);
        float og0 = sigmoid_fast(g[9]),  og1 = sigmoid_fast(g[10]), og2 = sigmoid_fast(g[11]);

        c0 = fg0 * c0 + ig0 * gg0;
        c1 = fg1 * c1 + ig1 * gg1;
        c2 = fg2 * c2 + ig2 * gg2;

        h0 = og0 * tanh_fast(c0);
        h1 = og1 * tanh_fast(c1);
        h2 = og2 * tanh_fast(c2);

        // ---- softmax over the 3 hidden units (numerically stable).
        float mx  = fmaxf(h0, fmaxf(h1, h2));
        float e0  = __expf(h0 - mx);
        float e1  = __expf(h1 - mx);
        float e2  = __expf(h2 - mx);
        float inv = rcp_fast(e0 + e1 + e2);

        if (lane == 0) {
            out[cell * HID + 0] = e0 * inv;
            out[cell * HID + 1] = e1 * inv;
            out[cell * HID + 2] = e2 * inv;
        }
        // loop backedge re-converges EXEC to all-1s before the next WMMA
    }
}

extern "C" void kernel_launch(void* const* d_in, const int* in_sizes, int n_in,
                              void* d_out, int out_size, void* d_ws, size_t ws_size,
                              hipStream_t stream) {
    (void)in_sizes; (void)n_in; (void)d_ws; (void)ws_size; (void)out_size;
    const float* state = (const float*)d_in[0];
    const float* W_ih  = (const float*)d_in[1];
    const float* W_hh  = (const float*)d_in[2];
    const float* b_ih  = (const float*)d_in[3];
    const float* b_hh  = (const float*)d_in[4];
    float* out = (float*)d_out;

    // One wave32: the whole problem is a sequential dependency chain.
    lstm_chain_wmma_kernel<<<1, 32, 0, stream>>>(state, W_ih, W_hh, b_ih, b_hh, out);
}